// OccupancyManager_29145648071306
// MI455X (gfx1250) — compile-verified
//
#include <hip/hip_runtime.h>

// Instant-NGP multiresolution hash encoding for MI455X (gfx1250, wave32).
// Gather-bound kernel: 16 levels x 8 random float2 gathers per point from
// 4 MB/level tables (64 MB total, resident in 192 MB L2). Output (256 MB) and
// xyz (24 MB) streams are non-temporal so they don't evict the tables.
// Software-pipelined across levels: 16 gathers in flight per wave
// (wait loadcnt<=8 instead of a full stall at every level).

typedef float v2f __attribute__((ext_vector_type(2)));
typedef float v4f __attribute__((ext_vector_type(4)));

#define NGP_LEVELS     16
#define NGP_TABLE_SIZE (1u << 19)
#define NGP_TABLE_MASK (NGP_TABLE_SIZE - 1u)
#define NGP_P1         2654435761u
#define NGP_P2         805459861u
#define NGP_BLOCK      256
#define NGP_ROWPAD     34   // 32 feats + 2 pad: 8B-aligned v2f rows, conflict-free

__device__ __forceinline__ v2f lerp2(v2f a, v2f b, float w) {
    return a + w * (b - a);            // fp-contract -> v_fma / v_pk_fma
}

// Hash-grid corner indices + trilinear fractions for one level.
__device__ __forceinline__ void ngp_indices(float ux, float uy, float uz, float res,
                                            unsigned idx[8],
                                            float& wx, float& wy, float& wz) {
    const float px = ux * res, py = uy * res, pz = uz * res;
    const float fx = floorf(px), fy = floorf(py), fz = floorf(pz);
    wx = px - fx; wy = py - fy; wz = pz - fz;

    const unsigned x0 = (unsigned)fx, y0 = (unsigned)fy, z0 = (unsigned)fz;
    // primes {1, P1, P2}; share partial products across the 8 corners
    const unsigned hx0 = x0,          hx1 = x0 + 1u;
    const unsigned hy0 = y0 * NGP_P1, hy1 = hy0 + NGP_P1;
    const unsigned hz0 = z0 * NGP_P2, hz1 = hz0 + NGP_P2;

    const unsigned h00 = hx0 ^ hy0, h10 = hx1 ^ hy0;
    const unsigned h01 = hx0 ^ hy1, h11 = hx1 ^ hy1;

    idx[0] = (h00 ^ hz0) & NGP_TABLE_MASK;
    idx[1] = (h10 ^ hz0) & NGP_TABLE_MASK;
    idx[2] = (h01 ^ hz0) & NGP_TABLE_MASK;
    idx[3] = (h11 ^ hz0) & NGP_TABLE_MASK;
    idx[4] = (h00 ^ hz1) & NGP_TABLE_MASK;
    idx[5] = (h10 ^ hz1) & NGP_TABLE_MASK;
    idx[6] = (h01 ^ hz1) & NGP_TABLE_MASK;
    idx[7] = (h11 ^ hz1) & NGP_TABLE_MASK;
}

__global__ __launch_bounds__(NGP_BLOCK)
void ngp_hash_encode_kernel(const float* __restrict__ xyz,
                            const float* __restrict__ tables,
                            const int*   __restrict__ resolutions,
                            float*       __restrict__ out,
                            int n_points)
{
    __shared__ __align__(16) float tile[NGP_BLOCK * NGP_ROWPAD];

    const unsigned tid = threadIdx.x;
    const unsigned p   = blockIdx.x * NGP_BLOCK + tid;
    const bool     live = (p < (unsigned)n_points);

    // ---- streaming (non-temporal) xyz load ----
    float x = 0.f, y = 0.f, z = 0.f;
    if (live) {
        const float* q = xyz + 3ull * p;
        x = __builtin_nontemporal_load(q + 0);
        y = __builtin_nontemporal_load(q + 1);
        z = __builtin_nontemporal_load(q + 2);
    }
    // u = xyz/SIZE + 0.5, SIZE = 2  (dead lanes get u=0.5 -> valid addresses)
    const float ux = x * 0.5f + 0.5f;
    const float uy = y * 0.5f + 0.5f;
    const float uz = z * 0.5f + 0.5f;

    const v2f* __restrict__ tb_base = (const v2f*)tables;

    // ---- pipeline prologue: issue level 0 gathers ----
    unsigned idx[8];
    float wx, wy, wz;
    ngp_indices(ux, uy, uz, (float)resolutions[0], idx, wx, wy, wz);
    v2f f[8];
    #pragma unroll
    for (int i = 0; i < 8; ++i) f[i] = tb_base[idx[i]];

    #pragma unroll
    for (int l = 0; l < NGP_LEVELS; ++l) {
        // issue next level's 8 gathers BEFORE consuming this level's
        v2f g[8];
        unsigned nidx[8];
        float nwx = 0.f, nwy = 0.f, nwz = 0.f;
        if (l + 1 < NGP_LEVELS) {
            ngp_indices(ux, uy, uz, (float)resolutions[l + 1], nidx, nwx, nwy, nwz);
            const v2f* __restrict__ tb = tb_base + (size_t)(l + 1) * NGP_TABLE_SIZE;
            #pragma unroll
            for (int i = 0; i < 8; ++i) g[i] = tb[nidx[i]];
        }

        // consume current level (in-order VMEM returns -> wait loadcnt<=8)
        const v2f a00 = lerp2(f[0], f[1], wx);
        const v2f a10 = lerp2(f[2], f[3], wx);
        const v2f a01 = lerp2(f[4], f[5], wx);
        const v2f a11 = lerp2(f[6], f[7], wx);
        const v2f b0  = lerp2(a00, a10, wy);
        const v2f b1  = lerp2(a01, a11, wy);
        const v2f e   = lerp2(b0, b1, wz);

        // single 8B-aligned ds_store_b64 per level (ROWPAD=34 -> even row stride)
        *reinterpret_cast<v2f*>(&tile[tid * NGP_ROWPAD + 2 * l]) = e;

        // rotate pipeline registers (free after full unroll)
        #pragma unroll
        for (int i = 0; i < 8; ++i) f[i] = g[i];
        wx = nwx; wy = nwy; wz = nwz;
    }

    __syncthreads();

    // ---- coalesced non-temporal writeback: block tile = contiguous
    //      256*32 f32 = 2048 float4 region of `out` ----
    v4f* __restrict__ outv = (v4f*)(out + (size_t)blockIdx.x * NGP_BLOCK * 32);
    const unsigned base_pt = blockIdx.x * NGP_BLOCK;

    #pragma unroll
    for (int j = 0; j < 8; ++j) {
        const unsigned idx4 = (unsigned)j * NGP_BLOCK + tid;  // float4 index 0..2047
        const unsigned fb   = idx4 * 4u;                      // logical float index
        const unsigned row  = fb >> 5;                        // point within block
        const unsigned col  = fb & 31u;                       // feat (multiple of 4)
        if (base_pt + row < (unsigned)n_points) {
            const float* s = &tile[row * NGP_ROWPAD + col];
            v4f v;
            v.x = s[0]; v.y = s[1]; v.z = s[2]; v.w = s[3];
            __builtin_nontemporal_store(v, outv + idx4);
        }
    }
}

extern "C" void kernel_launch(void* const* d_in, const int* in_sizes, int n_in,
                              void* d_out, int out_size, void* d_ws, size_t ws_size,
                              hipStream_t stream) {
    (void)n_in; (void)d_ws; (void)ws_size; (void)out_size;
    const float* xyz         = (const float*)d_in[0];
    const float* tables      = (const float*)d_in[1];
    const int*   resolutions = (const int*)d_in[2];
    float*       out         = (float*)d_out;

    const int n_points = in_sizes[0] / 3;                    // [N,3] flattened
    const int blocks   = (n_points + NGP_BLOCK - 1) / NGP_BLOCK;

    ngp_hash_encode_kernel<<<blocks, NGP_BLOCK, 0, stream>>>(
        xyz, tables, resolutions, out, n_points);
}